// Per_Region_Normalization_35639638622448
// MI455X (gfx1250) — compile-verified
//
#include <hip/hip_runtime.h>
#include <hip/hip_bf16.h>

typedef __attribute__((ext_vector_type(2))) float v2f;
typedef __attribute__((ext_vector_type(8))) float v8f;
typedef __attribute__((ext_vector_type(4))) int   v4i;

#define BB   8
#define CC   256
#define HH   128
#define WW   128
#define JJ   8
#define SS   256
#define HWSZ (HH*WW)
#define EPSV 1e-5f
#define ROWS 8   // rows per block in the final kernel

// gfx1250 async global->LDS copy (ASYNCcnt-tracked), guarded so the build
// cannot fail if the toolchain lacks the builtins.
#if defined(__has_builtin)
#if __has_builtin(__builtin_amdgcn_global_load_async_to_lds_b128) && \
    __has_builtin(__builtin_amdgcn_s_wait_asynccnt)
#define HAVE_ASYNC_LDS 1
#endif
#endif

#define AS1 __attribute__((address_space(1)))
#define AS3 __attribute__((address_space(3)))

// ---------------------------------------------------------------------------
// 1) BN statistics, stage 1: per-(channel,batch) partial sums (deterministic)
// ---------------------------------------------------------------------------
__global__ void bn_partial_k(const float* __restrict__ fp,
                             float* __restrict__ psum, float* __restrict__ psq) {
  const int c = blockIdx.x, b = blockIdx.y;
  const float* p = fp + ((size_t)(b*CC + c))*HWSZ;
  float s = 0.f, q = 0.f;
  for (int i = threadIdx.x; i < HWSZ; i += 256) { float v = p[i]; s += v; q += v*v; }
  __shared__ float ss[256], sq[256];
  ss[threadIdx.x] = s; sq[threadIdx.x] = q; __syncthreads();
  for (int off = 128; off > 0; off >>= 1) {
    if (threadIdx.x < off) { ss[threadIdx.x] += ss[threadIdx.x+off]; sq[threadIdx.x] += sq[threadIdx.x+off]; }
    __syncthreads();
  }
  if (threadIdx.x == 0) { psum[c*BB + b] = ss[0]; psq[c*BB + b] = sq[0]; }
}

// ---------------------------------------------------------------------------
// 2) BN finalize: scale[c] = w*rsqrt(var+eps), shift[c] = b - mean*scale
// ---------------------------------------------------------------------------
__global__ void bn_finalize_k(const float* __restrict__ psum, const float* __restrict__ psq,
                              const float* __restrict__ bnw, const float* __restrict__ bnb,
                              float* __restrict__ scale, float* __restrict__ shift) {
  const int c = threadIdx.x;
  float s = 0.f, q = 0.f;
  #pragma unroll
  for (int b = 0; b < BB; ++b) { s += psum[c*BB + b]; q += psq[c*BB + b]; }
  const float n    = (float)(BB*HWSZ);
  const float mean = s / n;
  const float var  = q / n - mean*mean;
  const float sc   = bnw[c] * rsqrtf(var + EPSV);
  scale[c] = sc;
  shift[c] = bnb[c] - mean*sc;
}

// ---------------------------------------------------------------------------
// 3) labels[b,h,w] = argmax_j sg[b,j,h,w]   (sg is one-hot)
// ---------------------------------------------------------------------------
__global__ void labels_k(const float* __restrict__ sg, int* __restrict__ labels) {
  const int idx = blockIdx.x*256 + threadIdx.x;      // 0 .. B*HW-1
  const int b = idx >> 14, pix = idx & (HWSZ-1);
  const float* p = sg + (size_t)b*JJ*HWSZ + pix;
  int lab = 0;
  #pragma unroll
  for (int j = 0; j < JJ; ++j) if (p[(size_t)j*HWSZ] > 0.5f) lab = j;
  labels[idx] = lab;
}

// ---------------------------------------------------------------------------
// 4) mu[b,j,t] = relu(fc_b[j,t] + sum_s fc_w[j,t,s] * code[b,j,s])
//    code selected: style_codes[b, mask?j:J, :]
// ---------------------------------------------------------------------------
__global__ void mu_k(const float* __restrict__ style, const int* __restrict__ mask,
                     const float* __restrict__ fcw, const float* __restrict__ fcb,
                     float* __restrict__ mu) {
  const int bj = blockIdx.x, b = bj >> 3, j = bj & 7;
  __shared__ float code[SS];
  const int sel = (mask[bj] == 1) ? j : JJ;
  code[threadIdx.x] = style[((size_t)b*(JJ+1) + sel)*SS + threadIdx.x];
  __syncthreads();
  const float* wr = fcw + ((size_t)(j*SS + threadIdx.x))*SS;
  float acc = fcb[j*SS + threadIdx.x];
  #pragma unroll 4
  for (int s = 0; s < SS; ++s) acc = fmaf(wr[s], code[s], acc);
  mu[(size_t)bj*SS + threadIdx.x] = fmaxf(acc, 0.f);
}

// ---------------------------------------------------------------------------
// 5) WMMA f32 16x16x4: G[conv][n=(b*8+j)][k][co] = sum_s Wconv[co,s,k]*mu[n,s]
//    One wave per 16x16 tile of (co x n); grid = 2 convs * 9 taps * 16 * 4.
//    Layout per CDNA5 ISA 7.12.2:
//      A 16x4 f32: lane m (0-15) v[r]=A[m, kk+r]; lanes 16-31 v[r]=A[m, kk+2+r]
//      B 4x16 f32: lane n        v[r]=B[kk+r, n]; lanes 16-31 v[r]=B[kk+2+r, n]
//      D 16x16:    vgpr i, lanes 0-15 -> M=i, lanes 16-31 -> M=8+i
// ---------------------------------------------------------------------------
__global__ void gtab_k(const float* __restrict__ wg, const float* __restrict__ wb,
                       const float* __restrict__ mu, float* __restrict__ G) {
  const int id   = blockIdx.x;
  const int conv = id / 576;           // 576 = 9*16*4
  const int rem  = id % 576;
  const int k    = rem >> 6;           // tap 0..8
  const int mt   = (rem >> 2) & 15;    // co tile
  const int nt   = rem & 3;            // (b,j) tile
  const float* w = conv ? wb : wg;

  const int lane = threadIdx.x;
  const int half = lane >> 4;          // K-half select
  const int lm   = lane & 15;

  const float* wbase = w + (size_t)(mt*16 + lm)*(SS*9) + k;   // [co][s][k], s-stride 9
  const float* mbase = mu + (size_t)(nt*16 + lm)*SS;

  v8f acc = {};
  for (int kk = 0; kk < SS; kk += 4) {
    const int s0 = kk + 2*half;
    v2f a, bv;
    a.x  = wbase[(size_t)s0*9];
    a.y  = wbase[(size_t)(s0+1)*9];
    bv.x = mbase[s0];
    bv.y = mbase[s0+1];
    acc = __builtin_amdgcn_wmma_f32_16x16x4_f32(false, a, false, bv,
                                                (short)0, acc, false, false);
  }
  // D element (m = i + 8*half, n = lm)
  float* out = G + ((((size_t)conv*64 + (nt*16 + lm))*9 + k)*CC) + mt*16;
  #pragma unroll
  for (int i = 0; i < 8; ++i) out[i + 8*half] = acc[i];
}

// ---------------------------------------------------------------------------
// 6) Fused final kernel: out = xn*(1+gamma)+beta with
//    gamma[b,co,h,w] = gbias[co] + sum_k G[0][b,r_k][k][co]  (r_k from labels,
//    sentinel region 8 == zero handles SAME zero-padding), beta analogous.
//    G[b] (+ zero region) staged in LDS (~166 KB of the 320 KB WGP LDS) via
//    CDNA5 async global->LDS b128 copies (ASYNCcnt) when available.
//    Threads map to w (coalesced HBM), co-loop unrolled x4 -> ds_load_b128.
// ---------------------------------------------------------------------------
#define GREG ((JJ+1)*9*CC)   // 20736 floats per conv table incl. zero region

__global__ void final_k(const float* __restrict__ fp, const int* __restrict__ labels,
                        const float* __restrict__ G,
                        const float* __restrict__ scale, const float* __restrict__ shift,
                        const float* __restrict__ gbias, const float* __restrict__ bbias,
                        float* __restrict__ out) {
  extern __shared__ float smem[];
  float* ldsG   = smem;                  // GREG
  float* ldsB   = ldsG + GREG;           // GREG
  float* sScale = ldsB + GREG;           // 256
  float* sShift = sScale + CC;           // 256
  float* sGb    = sShift + CC;           // 256
  float* sBb    = sGb + CC;              // 256
  int*   sLab   = (int*)(sBb + CC);      // (ROWS+2)*W = 1280

  const int b  = blockIdx.x >> 4;
  const int h0 = (blockIdx.x & 15) * ROWS;
  const int t  = threadIdx.x;

  // stage G/B tables for this batch image (regions 0..7), zero region 8
  const float* gsrc = G + ((size_t)0*64 + b*8)*9*CC;
  const float* bsrc = G + ((size_t)1*64 + b*8)*9*CC;
#if defined(HAVE_ASYNC_LDS)
  // async global->LDS, 4 floats per lane per issue, no VGPR round-trip
  for (int i = 4*t; i < 8*9*CC; i += 4*256) {
    __builtin_amdgcn_global_load_async_to_lds_b128(
        (AS1 v4i*)(gsrc + i), (AS3 v4i*)(ldsG + i), 0, 0);
    __builtin_amdgcn_global_load_async_to_lds_b128(
        (AS1 v4i*)(bsrc + i), (AS3 v4i*)(ldsB + i), 0, 0);
  }
#else
  for (int i = t; i < 8*9*CC; i += 256) { ldsG[i] = gsrc[i]; ldsB[i] = bsrc[i]; }
#endif
  for (int i = t; i < 9*CC;   i += 256) { ldsG[8*9*CC + i] = 0.f; ldsB[8*9*CC + i] = 0.f; }
  sScale[t] = scale[t]; sShift[t] = shift[t]; sGb[t] = gbias[t]; sBb[t] = bbias[t];
  for (int i = t; i < (ROWS+2)*WW; i += 256) {
    const int r = i / WW, w = i % WW, h = h0 - 1 + r;
    sLab[i] = (h >= 0 && h < HH) ? labels[(size_t)b*HWSZ + h*WW + w] : 8;
  }
#if defined(HAVE_ASYNC_LDS)
  __builtin_amdgcn_s_wait_asynccnt(0);   // this wave's async LDS writes done
#endif
  __syncthreads();

  #pragma unroll
  for (int pp = 0; pp < (ROWS*WW)/256; ++pp) {
    const int p  = pp*256 + t;
    const int hh = p >> 7;          // 0..ROWS-1
    const int w  = p & 127;

    int off[9];
    #pragma unroll
    for (int dh = 0; dh < 3; ++dh)
      #pragma unroll
      for (int dw = 0; dw < 3; ++dw) {
        const int ww2 = w + dw - 1;
        const int lab = (ww2 >= 0 && ww2 < WW) ? sLab[(hh+dh)*WW + ww2] : 8;
        off[3*dh+dw] = (lab*9 + (3*dh+dw)) * CC;
      }

    const float* fpb  = fp  + ((size_t)b*CC)*HWSZ + (h0+hh)*WW + w;
    float*       outb = out + ((size_t)b*CC)*HWSZ + (h0+hh)*WW + w;

    for (int co = 0; co < CC; co += 4) {
      float ga[4], be[4];
      {
        const float4 g0 = *(const float4*)&sGb[co];
        const float4 b0 = *(const float4*)&sBb[co];
        ga[0]=g0.x; ga[1]=g0.y; ga[2]=g0.z; ga[3]=g0.w;
        be[0]=b0.x; be[1]=b0.y; be[2]=b0.z; be[3]=b0.w;
      }
      #pragma unroll
      for (int k = 0; k < 9; ++k) {
        const float4 gv = *(const float4*)&ldsG[off[k] + co];   // ds_load_b128
        const float4 bv = *(const float4*)&ldsB[off[k] + co];
        ga[0]+=gv.x; ga[1]+=gv.y; ga[2]+=gv.z; ga[3]+=gv.w;
        be[0]+=bv.x; be[1]+=bv.y; be[2]+=bv.z; be[3]+=bv.w;
      }
      #pragma unroll
      for (int u = 0; u < 4; ++u) {
        const int cc = co + u;
        const float f  = fpb[(size_t)cc*HWSZ];
        const float xn = fmaf(f, sScale[cc], sShift[cc]);
        outb[(size_t)cc*HWSZ] = fmaf(xn, ga[u], xn) + be[u];   // xn*(1+g)+beta
      }
    }
  }
}

// ---------------------------------------------------------------------------
// launcher
// ---------------------------------------------------------------------------
extern "C" void kernel_launch(void* const* d_in, const int* in_sizes, int n_in,
                              void* d_out, int out_size, void* d_ws, size_t ws_size,
                              hipStream_t stream) {
  const float* fp     = (const float*)d_in[0];
  const float* sg     = (const float*)d_in[1];
  const float* style  = (const float*)d_in[2];
  const int*   maskc  = (const int*)  d_in[3];
  const float* bnw    = (const float*)d_in[4];
  const float* bnb    = (const float*)d_in[5];
  const float* cgw    = (const float*)d_in[6];
  const float* cgb    = (const float*)d_in[7];
  const float* cbw    = (const float*)d_in[8];
  const float* cbb    = (const float*)d_in[9];
  const float* fcw    = (const float*)d_in[10];
  const float* fcb    = (const float*)d_in[11];
  float*       out    = (float*)d_out;

  // workspace carve (floats)
  float* ws    = (float*)d_ws;
  float* psum  = ws;               // 2048
  float* psq   = ws + 2048;        // 2048
  float* scale = ws + 4096;        // 256
  float* shift = ws + 4352;        // 256
  float* mu    = ws + 4608;        // 64*256   = 16384
  float* G     = ws + 20992;       // 2*64*9*256 = 294912
  int*   labels= (int*)(ws + 315904); // 8*16384 ints

  bn_partial_k<<<dim3(CC, BB), 256, 0, stream>>>(fp, psum, psq);
  bn_finalize_k<<<1, 256, 0, stream>>>(psum, psq, bnw, bnb, scale, shift);
  labels_k<<<(BB*HWSZ)/256, 256, 0, stream>>>(sg, labels);
  mu_k<<<BB*JJ, 256, 0, stream>>>(style, maskc, fcw, fcb, mu);
  gtab_k<<<2*9*16*4, 32, 0, stream>>>(cgw, cbw, mu, G);

  const size_t smem = (size_t)(2*GREG + 4*CC)*sizeof(float) + (size_t)((ROWS+2)*WW)*sizeof(int);
  (void)hipFuncSetAttribute((const void*)final_k,
                            hipFuncAttributeMaxDynamicSharedMemorySize, (int)smem);
  final_k<<<BB*(HH/ROWS), 256, smem, stream>>>(fp, labels, G, scale, shift, cgb, cbb, out);
}